// BoundaryLoss_39951785787529
// MI455X (gfx1250) — compile-verified
//
#include <hip/hip_runtime.h>
#include <hip/hip_bf16.h>

// Problem constants (from reference): B=8,C=21,H=W=512, D=128,h=w=128, MEM=1000
#define NPIX   131072      // 8*128*128 flattened (b,y,x) samples
#define NBLK   512         // NPIX / 256
#define HW     262144      // 512*512
#define CHW    (21 * 262144)
#define EMB_B  (128 * 16384)   // embeddings b-stride (D*h*w)
#define EMB_D  16384           // embeddings d-stride (h*w)
#define KANC   100             // MEM//10
#define KPN    333             // MEM//3
#define SELCAP 352             // padded list capacity per category

typedef __attribute__((ext_vector_type(2))) float v2f;
typedef __attribute__((ext_vector_type(8))) float v8f;

// ---------------------------------------------------------------------------
// Stage 1: classify each of the 131072 downsampled pixels into
// anchor / positive / negative (flags bitmask) + per-block category counts.
// nearest-downsample 512->128 == sample (4y, 4x). pred_mask = argmax_c != 0
// == exists c>=1 with preds[c] > preds[0] (argmax keeps first max).
// ---------------------------------------------------------------------------
__global__ void k_classify(const float* __restrict__ preds,
                           const int* __restrict__ fsss,
                           unsigned char* __restrict__ flags,
                           int* __restrict__ bcnt) {
  __shared__ int cnt[3];
  int t = threadIdx.x;
  if (t < 3) cnt[t] = 0;
  __syncthreads();

  int n = blockIdx.x * 256 + t;
  int b = n >> 14, rem = n & 16383;
  int y = rem >> 7, x = rem & 127;
  size_t pix = (size_t)(y << 2) * 512 + (size_t)(x << 2);

  const float* pb = preds + (size_t)b * CHW + pix;
  float p0 = pb[0];
  bool pm = false;
#pragma unroll
  for (int c = 1; c < 21; ++c) pm = pm || (pb[(size_t)c * HW] > p0);

  int fv = fsss[(size_t)b * HW + pix];
  bool fm = (fv != 0) && (fv != 255);

  unsigned fa = (pm && fm) ? 1u : 0u;          // anchor
  unsigned fp = (fm && !pm) ? 1u : 0u;         // positive
  unsigned fn = (pm && (fv == 0)) ? 1u : 0u;   // negative
  flags[n] = (unsigned char)(fa | (fp << 1) | (fn << 2));

  unsigned long long ba = __ballot(fa);
  unsigned long long bp = __ballot(fp);
  unsigned long long bn = __ballot(fn);
  if ((t & 31) == 0) {
    atomicAdd(&cnt[0], __popcll(ba));
    atomicAdd(&cnt[1], __popcll(bp));
    atomicAdd(&cnt[2], __popcll(bn));
  }
  __syncthreads();
  if (t < 3) bcnt[t * NBLK + blockIdx.x] = cnt[t];
}

// ---------------------------------------------------------------------------
// Stage 2: exclusive scan of the 512 per-block counts (one serial thread per
// category -- 512 adds, negligible) + category totals.
// ---------------------------------------------------------------------------
__global__ void k_scan(const int* __restrict__ bcnt, int* __restrict__ bbase,
                       int* __restrict__ totals) {
  int cat = threadIdx.x;
  if (cat >= 3) return;
  int s = 0;
  for (int i = 0; i < NBLK; ++i) {
    bbase[cat * NBLK + i] = s;
    s += bcnt[cat * NBLK + i];
  }
  totals[cat] = s;
}

// ---------------------------------------------------------------------------
// Stage 3: global rank per pixel via block base + intra-block ballot prefix
// (wave32). First-k selection: rank < K -> selList[cat][rank] = n.
// Ranks are unique so this is atomic-free and fully deterministic.
// ---------------------------------------------------------------------------
__global__ void k_select(const unsigned char* __restrict__ flags,
                         const int* __restrict__ bbase,
                         int* __restrict__ selList) {
  __shared__ int wcnt[3][8];
  int t = threadIdx.x, wave = t >> 5, lane = t & 31;
  int n = blockIdx.x * 256 + t;
  unsigned f = flags[n];
  unsigned long long bal0 = __ballot(f & 1u);
  unsigned long long bal1 = __ballot((f >> 1) & 1u);
  unsigned long long bal2 = __ballot((f >> 2) & 1u);
  if (lane == 0) {
    wcnt[0][wave] = __popcll(bal0);
    wcnt[1][wave] = __popcll(bal1);
    wcnt[2][wave] = __popcll(bal2);
  }
  __syncthreads();
  unsigned long long lt = (1ull << lane) - 1ull;
  unsigned long long bals[3] = {bal0, bal1, bal2};
  for (int c = 0; c < 3; ++c) {
    if ((f >> c) & 1u) {
      int r = bbase[c * NBLK + blockIdx.x];
      for (int w = 0; w < wave; ++w) r += wcnt[c][w];
      r += __popcll(bals[c] & lt);
      int K = (c == 0) ? KANC : KPN;
      if (r < K) selList[c * SELCAP + r] = n;
    }
  }
}

// ---------------------------------------------------------------------------
// Stage 4: per selected pixel (<= 766 total), one wave computes
// 1 / max(||emb[:,y,x]||, 1e-12) via a 4-elem/lane strided gather + shuffle
// reduction.
// ---------------------------------------------------------------------------
__global__ void k_norms(const float* __restrict__ emb,
                        const int* __restrict__ selList,
                        const int* __restrict__ totals,
                        float* __restrict__ invn) {
  int w = blockIdx.x;             // 0..1055, blockDim = 32
  int cat = w / SELCAP, r = w % SELCAP;
  int K = (cat == 0) ? KANC : KPN;
  int tot = totals[cat];
  int Kc = tot < K ? tot : K;
  if (r >= Kc) return;
  int n = selList[cat * SELCAP + r];
  int b = n >> 14, rem = n & 16383;
  int y = rem >> 7, x = rem & 127;
  const float* base = emb + (size_t)b * EMB_B + (size_t)y * 128 + x;
  int lane = threadIdx.x;
  float ss = 0.f;
#pragma unroll
  for (int j = 0; j < 4; ++j) {
    float v = base[(size_t)(lane + 32 * j) * EMB_D];
    ss += v * v;
  }
  for (int off = 16; off > 0; off >>= 1) ss += __shfl_xor(ss, off, 32);
  if (lane == 0) invn[cat * SELCAP + r] = 1.0f / fmaxf(sqrtf(ss), 1e-12f);
}

// ---------------------------------------------------------------------------
// Stage 5: deterministic accumulation of the three 128-d sum vectors:
// acc[cat][d] = sum_{r<Kc} emb[d, sel_r] * invn_r   (fixed rank order).
// ---------------------------------------------------------------------------
__global__ void k_accum(const float* __restrict__ emb,
                        const int* __restrict__ selList,
                        const float* __restrict__ invn,
                        const int* __restrict__ totals,
                        float* __restrict__ acc) {
  int cat = blockIdx.x, d = threadIdx.x;   // 3 blocks x 128 threads
  int K = (cat == 0) ? KANC : KPN;
  int tot = totals[cat];
  int Kc = tot < K ? tot : K;
  float s = 0.f;
  for (int r = 0; r < Kc; ++r) {
    int n = selList[cat * SELCAP + r];
    int b = n >> 14, rem = n & 16383;
    int y = rem >> 7, x = rem & 127;
    float v = emb[(size_t)(b * 128 + d) * EMB_D + (size_t)y * 128 + x];
    s += v * invn[cat * SELCAP + r];
  }
  acc[cat * 128 + d] = s;
}

// ---------------------------------------------------------------------------
// Stage 6: memory-bank tail column sums via V_WMMA_F32_16X16X4_F32
// (A = 16x4 all-ones, B = 4 rows x 16 cols of the bank, 250 accumulating
// WMMAs per tile; rows < skip contribute 0 via branchless select so EXEC
// stays all-ones), then dot products + final relu scalar.
// ---------------------------------------------------------------------------
__global__ void k_final(const float* __restrict__ pos_mem,
                        const float* __restrict__ neg_mem,
                        const float* __restrict__ acc,
                        const int* __restrict__ totals,
                        float* __restrict__ out) {
  __shared__ float tails[256];   // [0..127]=pos tail, [128..255]=neg tail
  __shared__ float red[128];
  int t = threadIdx.x;           // 512 threads = 16 waves
  int wave = t >> 5, lane = t & 31;

  int totA = totals[0], totP = totals[1], totN = totals[2];
  int Kp = totP < KPN ? totP : KPN;
  int Kn = totN < KPN ? totN : KPN;
  int nA = totA < KANC ? totA : KANC;
  float n_anc = fmaxf((float)nA, 1.0f);

  bool isNeg = wave >= 8;
  const float* mem = isNeg ? neg_mem : pos_mem;
  int skip = isNeg ? Kn : Kp;
  int col = ((wave & 7) << 4) + (lane & 15);   // lane%16 -> column N (B & D)
  int koff = (lane < 16) ? 0 : 2;

  v8f c = {0.f, 0.f, 0.f, 0.f, 0.f, 0.f, 0.f, 0.f};
  v2f a;
  a.x = 1.0f;
  a.y = 1.0f;                                  // A = all-ones 16x4
  for (int r0 = 0; r0 < 1000; r0 += 4) {
    int ra = r0 + koff;
    int rb = ra + 1;
    float va = mem[(size_t)ra * 128 + col];
    float vb = mem[(size_t)rb * 128 + col];
    v2f bm;
    bm.x = (ra >= skip) ? va : 0.0f;
    bm.y = (rb >= skip) ? vb : 0.0f;
    c = __builtin_amdgcn_wmma_f32_16x16x4_f32(false, a, false, bm, (short)0, c,
                                              false, false);
  }
  if (lane < 16) tails[(isNeg ? 128 : 0) + col] = c[0];  // D row M=0
  __syncthreads();

  if (t < 128) {
    float av = acc[t];                       // anchor sum
    float pv = acc[128 + t] + tails[t];      // pos bank column sum
    float nv = acc[256 + t] + tails[128 + t];// neg bank column sum
    red[t] = av * (pv - nv);
  }
  __syncthreads();
  for (int s = 64; s > 0; s >>= 1) {
    if (t < s) red[t] += red[t + s];
    __syncthreads();
  }
  if (t == 0) {
    float v = red[0] / (n_anc * 1000.0f) + 0.2f;
    out[0] = v > 0.0f ? v : 0.0f;
  }
}

// ---------------------------------------------------------------------------
// Workspace layout (bytes):
//   0      acc[3*128] f32          (written by k_accum)
//   1536   totals[3] int (+pad)
//   1552   bcnt[3*512] int
//   7696   bbase[3*512] int
//   13840  selList[3*352] int
//   18064  invn[3*352] f32
//   22288  flags[131072] u8        -> total ~150 KB
// ---------------------------------------------------------------------------
extern "C" void kernel_launch(void* const* d_in, const int* in_sizes, int n_in,
                              void* d_out, int out_size, void* d_ws, size_t ws_size,
                              hipStream_t stream) {
  const float* preds = (const float*)d_in[0];
  const float* emb   = (const float*)d_in[1];
  // d_in[2] = gts (unused by the loss)
  const int* fsss    = (const int*)d_in[3];
  const float* pos_m = (const float*)d_in[4];
  const float* neg_m = (const float*)d_in[5];

  char* ws = (char*)d_ws;
  float* acc            = (float*)(ws + 0);
  int* totals           = (int*)(ws + 1536);
  int* bcnt             = (int*)(ws + 1552);
  int* bbase            = (int*)(ws + 7696);
  int* selList          = (int*)(ws + 13840);
  float* invn           = (float*)(ws + 18064);
  unsigned char* flags  = (unsigned char*)(ws + 22288);

  k_classify<<<NBLK, 256, 0, stream>>>(preds, fsss, flags, bcnt);
  k_scan<<<1, 32, 0, stream>>>(bcnt, bbase, totals);
  k_select<<<NBLK, 256, 0, stream>>>(flags, bbase, selList);
  k_norms<<<3 * SELCAP, 32, 0, stream>>>(emb, selList, totals, invn);
  k_accum<<<3, 128, 0, stream>>>(emb, selList, invn, totals, acc);
  k_final<<<1, 512, 0, stream>>>(pos_m, neg_m, acc, totals, (float*)d_out);
}